// TripletLoss_85830626443972
// MI455X (gfx1250) — compile-verified
//
#include <hip/hip_runtime.h>
#include <math.h>

typedef __attribute__((ext_vector_type(2))) float v2f;
typedef __attribute__((ext_vector_type(8))) float v8f;

#define NROWS 4096
#define DIM   1024
#define BT    128            // tile edge (rows & cols) per block
#define NJB   (NROWS / BT)   // 32 column blocks
#define KC    16             // K chunk staged in LDS (double buffered)
#define NCHUNK (DIM / KC)    // 64 chunks
#define STRD  20             // LDS row stride in floats (80B rows: 16B-aligned, conflict-free)

__device__ __forceinline__ void wait_asynccnt0() {
#if __has_builtin(__builtin_amdgcn_s_wait_asynccnt)
    __builtin_amdgcn_s_wait_asynccnt(0);
#else
    asm volatile("s_wait_asynccnt 0x0" ::: "memory");
#endif
}

__device__ __forceinline__ void async_load_b128_to_lds(const float* gptr, const float* lptr) {
    // VDST VGPR holds LDS byte offset (generic shared addr low 32 bits == LDS offset),
    // VADDR pair holds the 64-bit global address; saddr = off (null).
    const unsigned int  loff = (unsigned int)(size_t)lptr;
    const unsigned long long ga = (unsigned long long)(size_t)gptr;
    asm volatile("global_load_async_to_lds_b128 %0, %1, off"
                 :: "v"(loff), "v"(ga) : "memory");
}

// ---------------- Kernel 1: L2-normalize rows (torch F.normalize w/ eps) ----------------
__global__ __launch_bounds__(256) void k_normalize(const float* __restrict__ X,
                                                   float* __restrict__ Nrm) {
    const int row = blockIdx.x;
    const int tid = threadIdx.x;
    const float4 x = ((const float4*)(X + (size_t)row * DIM))[tid];
    float s = x.x * x.x + x.y * x.y + x.z * x.z + x.w * x.w;
#pragma unroll
    for (int m = 16; m >= 1; m >>= 1) s += __shfl_xor(s, m, 32);
    __shared__ float red[8];
    const int wv = tid >> 5, ln = tid & 31;
    if (ln == 0) red[wv] = s;
    __syncthreads();
    if (tid == 0) {
        float t = 0.f;
#pragma unroll
        for (int w = 0; w < 8; ++w) t += red[w];
        red[0] = fmaxf(sqrtf(t), 1e-12f);
    }
    __syncthreads();
    const float inv = 1.0f / red[0];
    float4 y;
    y.x = x.x * inv; y.y = x.y * inv; y.z = x.z * inv; y.w = x.w * inv;
    ((float4*)(Nrm + (size_t)row * DIM))[tid] = y;
}

// ------- Kernel 2: S = N*N^T via V_WMMA_F32_16X16X4_F32 with async-LDS double buffering,
//         fused per-row online-softmax stats (max/argmax, min/argmin, sum-exp) -------
__global__ __launch_bounds__(256) void k_simstats(const float* __restrict__ Nrm,
                                                  float* __restrict__ pM,
                                                  float* __restrict__ pZ,
                                                  float* __restrict__ pMn,
                                                  int* __restrict__ pAmax,
                                                  int* __restrict__ pAmin) {
    __shared__ __align__(16) float As[2][BT * STRD];
    __shared__ __align__(16) float Bs[2][BT * STRD];
    const int jb = blockIdx.x, ib = blockIdx.y;
    const int i_base = ib * BT, j_base = jb * BT;
    const int tid  = threadIdx.x;
    const int lane = tid & 31, wave = tid >> 5;
    const int l16  = lane & 15, lh  = lane >> 4;

    // staging map: 128 rows x 16 floats = 512 float4 per array; 2 per thread per array
    const int srow0 = tid >> 2;                 // rows 0..63   (r = 0)
    const int srow1 = (tid + 256) >> 2;         // rows 64..127 (r = 1)
    const int sc4   = (tid & 3) * 4;

    v8f acc[8];
#pragma unroll
    for (int t = 0; t < 8; ++t) acc[t] = (v8f){0.f, 0.f, 0.f, 0.f, 0.f, 0.f, 0.f, 0.f};

    // ---- prologue: stage chunk 0 into buffer 0 ----
    async_load_b128_to_lds(Nrm + (size_t)(i_base + srow0) * DIM + sc4, &As[0][srow0 * STRD + sc4]);
    async_load_b128_to_lds(Nrm + (size_t)(j_base + srow0) * DIM + sc4, &Bs[0][srow0 * STRD + sc4]);
    async_load_b128_to_lds(Nrm + (size_t)(i_base + srow1) * DIM + sc4, &As[0][srow1 * STRD + sc4]);
    async_load_b128_to_lds(Nrm + (size_t)(j_base + srow1) * DIM + sc4, &Bs[0][srow1 * STRD + sc4]);

    for (int c = 0; c < NCHUNK; ++c) {
        const int buf = c & 1;
        wait_asynccnt0();        // own stage-c ops complete (stage c+1 not issued yet)
        __syncthreads();         // all waves: stage c visible; done reading buf^1 from chunk c-1
        if (c + 1 < NCHUNK) {
            const int nb = buf ^ 1;
            const int kc = (c + 1) * KC;
            async_load_b128_to_lds(Nrm + (size_t)(i_base + srow0) * DIM + kc + sc4,
                                   &As[nb][srow0 * STRD + sc4]);
            async_load_b128_to_lds(Nrm + (size_t)(j_base + srow0) * DIM + kc + sc4,
                                   &Bs[nb][srow0 * STRD + sc4]);
            async_load_b128_to_lds(Nrm + (size_t)(i_base + srow1) * DIM + kc + sc4,
                                   &As[nb][srow1 * STRD + sc4]);
            async_load_b128_to_lds(Nrm + (size_t)(j_base + srow1) * DIM + kc + sc4,
                                   &Bs[nb][srow1 * STRD + sc4]);
        }
        // A-frag (16x4 f32): lanes 0-15 -> row m, K={0,1}; lanes 16-31 -> K={2,3}.
        // S symmetric => B-frag layout identical with j-strip rows.
#pragma unroll
        for (int kb = 0; kb < KC; kb += 4) {
            const v2f a = *(const v2f*)&As[buf][(16 * wave + l16) * STRD + kb + 2 * lh];
#pragma unroll
            for (int jt = 0; jt < 8; ++jt) {
                const v2f b = *(const v2f*)&Bs[buf][(16 * jt + l16) * STRD + kb + 2 * lh];
                acc[jt] = __builtin_amdgcn_wmma_f32_16x16x4_f32(
                    /*neg_a=*/false, a, /*neg_b=*/false, b,
                    /*c_mod=*/(short)0, acc[jt], /*reuse_a=*/false, /*reuse_b=*/false);
            }
        }
    }

    // C layout: acc[jt][v] at lane L = S[i_base+16*wave+8*(L/16)+v][j_base+16*jt+(L%16)]
#pragma unroll
    for (int v = 0; v < 8; ++v) {
        const int i = i_base + 16 * wave + 8 * lh + v;
        float m  = -__builtin_inff(); int am = 0x7fffffff;
        float mn =  __builtin_inff(); int an = 0x7fffffff;
        float Z  = 0.0f;
#pragma unroll
        for (int jt = 0; jt < 8; ++jt) {
            const int j = j_base + 16 * jt + l16;
            if (i == j) continue;                    // off-diagonal only
            const float s = acc[jt][v];
            if (s > m) { Z = Z * expf(m - s) + 1.0f; m = s; am = j; }
            else       { Z += expf(s - m); if (s == m && j < am) am = j; }
            if (s < mn)                  { mn = s; an = j; }
            else if (s == mn && j < an)  { an = j; }
        }
        // butterfly across the 16 lanes holding the same row (masks stay inside half-wave)
#pragma unroll
        for (int msk = 1; msk < 16; msk <<= 1) {
            const float m2  = __shfl_xor(m,  msk, 32);
            const int   am2 = __shfl_xor(am, msk, 32);
            const float mn2 = __shfl_xor(mn, msk, 32);
            const int   an2 = __shfl_xor(an, msk, 32);
            const float Z2  = __shfl_xor(Z,  msk, 32);
            const float mx  = fmaxf(m, m2);
            Z = Z * expf(m - mx) + Z2 * expf(m2 - mx);
            if (m2 > m || (m2 == m && am2 < am)) am = am2;
            m = mx;
            if (mn2 < mn || (mn2 == mn && an2 < an)) { mn = mn2; an = an2; }
        }
        if (l16 == 0) {
            const size_t o = (size_t)i * NJB + jb;
            pM[o] = m; pZ[o] = Z; pMn[o] = mn; pAmax[o] = am; pAmin[o] = an;
        }
    }
}

// ---------------- Kernel 3: merge per-row partials -> most/least/margin ----------------
__global__ __launch_bounds__(256) void k_combine(const float* __restrict__ pM,
                                                 const float* __restrict__ pZ,
                                                 const float* __restrict__ pMn,
                                                 const int* __restrict__ pAmax,
                                                 const int* __restrict__ pAmin,
                                                 int* __restrict__ most,
                                                 int* __restrict__ least,
                                                 float* __restrict__ margin,
                                                 float* __restrict__ out) {
    const int row = blockIdx.x * 256 + threadIdx.x;
    if (row == 0) out[0] = 0.0f;                     // zero accumulator before k_loss
    if (row >= NROWS) return;
    float m  = -__builtin_inff(); int am = 0x7fffffff;
    float mn =  __builtin_inff(); int an = 0x7fffffff;
    float Z  = 0.0f;
    for (int b = 0; b < NJB; ++b) {
        const size_t o = (size_t)row * NJB + b;
        const float m2 = pM[o], Z2 = pZ[o], mn2 = pMn[o];
        const int am2 = pAmax[o], an2 = pAmin[o];
        const float mx = fmaxf(m, m2);
        Z = Z * expf(m - mx) + Z2 * expf(m2 - mx);
        if (m2 > m || (m2 == m && am2 < am)) am = am2;
        m = mx;
        if (mn2 < mn || (mn2 == mn && an2 < an)) { mn = mn2; an = an2; }
    }
    most[row]  = am;
    least[row] = an;
    // s_most = 1/Z ; s_least = exp(mn - m)/Z  (softmax with diag masked to -inf)
    margin[row] = (1.0f - expf(mn - m)) / Z;
}

// ---------------- Kernel 4: distances + ReLU margin, mean into d_out ----------------
__global__ __launch_bounds__(256) void k_loss(const float* __restrict__ P,
                                              const int* __restrict__ most,
                                              const int* __restrict__ least,
                                              const float* __restrict__ margin,
                                              float* __restrict__ out) {
    const int i = blockIdx.x;
    const int tid = threadIdx.x;
    const int mo = most[i], le = least[i];
    const float4 a = ((const float4*)(P + (size_t)i  * DIM))[tid];
    const float4 b = ((const float4*)(P + (size_t)mo * DIM))[tid];
    const float4 c = ((const float4*)(P + (size_t)le * DIM))[tid];
    float ds = (a.x - b.x) * (a.x - b.x) + (a.y - b.y) * (a.y - b.y) +
               (a.z - b.z) * (a.z - b.z) + (a.w - b.w) * (a.w - b.w);
    float dd = (a.x - c.x) * (a.x - c.x) + (a.y - c.y) * (a.y - c.y) +
               (a.z - c.z) * (a.z - c.z) + (a.w - c.w) * (a.w - c.w);
#pragma unroll
    for (int m = 16; m >= 1; m >>= 1) {
        ds += __shfl_xor(ds, m, 32);
        dd += __shfl_xor(dd, m, 32);
    }
    __shared__ float rs[8], rd[8];
    const int wv = tid >> 5, ln = tid & 31;
    if (ln == 0) { rs[wv] = ds; rd[wv] = dd; }
    __syncthreads();
    if (tid == 0) {
        float t1 = 0.f, t2 = 0.f;
#pragma unroll
        for (int w = 0; w < 8; ++w) { t1 += rs[w]; t2 += rd[w]; }
        const float loss = fmaxf(0.0f, sqrtf(t1) - sqrtf(t2) + margin[i]);
        atomicAdd(out, loss * (1.0f / (float)NROWS));
    }
}

extern "C" void kernel_launch(void* const* d_in, const int* in_sizes, int n_in,
                              void* d_out, int out_size, void* d_ws, size_t ws_size,
                              hipStream_t stream) {
    const float* SE = (const float*)d_in[0];   // semantic_embeddings [4096,1024] f32
    const float* P  = (const float*)d_in[1];   // projected_embeddings [4096,1024] f32
    float* out = (float*)d_out;

    char* ws = (char*)d_ws;
    size_t off = 0;
    float* Nrm   = (float*)(ws + off); off += (size_t)NROWS * DIM * sizeof(float);   // 16 MB
    float* pM    = (float*)(ws + off); off += (size_t)NROWS * NJB * sizeof(float);
    float* pZ    = (float*)(ws + off); off += (size_t)NROWS * NJB * sizeof(float);
    float* pMn   = (float*)(ws + off); off += (size_t)NROWS * NJB * sizeof(float);
    int*   pAmax = (int*)  (ws + off); off += (size_t)NROWS * NJB * sizeof(int);
    int*   pAmin = (int*)  (ws + off); off += (size_t)NROWS * NJB * sizeof(int);
    int*   most  = (int*)  (ws + off); off += (size_t)NROWS * sizeof(int);
    int*   least = (int*)  (ws + off); off += (size_t)NROWS * sizeof(int);
    float* margv = (float*)(ws + off); off += (size_t)NROWS * sizeof(float);

    k_normalize<<<NROWS, 256, 0, stream>>>(SE, Nrm);

    dim3 g2(NJB, NROWS / BT);
    k_simstats<<<g2, 256, 0, stream>>>(Nrm, pM, pZ, pMn, pAmax, pAmin);

    k_combine<<<NROWS / 256, 256, 0, stream>>>(pM, pZ, pMn, pAmax, pAmin,
                                               most, least, margv, out);

    k_loss<<<NROWS, 256, 0, stream>>>(P, most, least, margv, out);
}